// GCN_9517647528627
// MI455X (gfx1250) — compile-verified
//
#include <hip/hip_runtime.h>
#include <hip/hip_bf16.h>

// Problem constants (match reference)
#define N_NODES   50000
#define N_EDGES   800000
#define N_GRAPHS  2048
#define IN_FEATS  59
#define HDIM      256
#define OUTC      2
#define L_CONV    3
#define L_FC      3

typedef __attribute__((ext_vector_type(2))) float v2f;
typedef __attribute__((ext_vector_type(8))) float v8f;

// ---------------------------------------------------------------- small utils
__global__ void deg_init_k(float* __restrict__ deg, int n) {
    int i = blockIdx.x * blockDim.x + threadIdx.x;
    if (i < n) deg[i] = 1.0f;                 // self-loop counted
}

__global__ void deg_count_k(const int* __restrict__ dst, float* __restrict__ deg, int e) {
    int i = blockIdx.x * blockDim.x + threadIdx.x;
    if (i < e) atomicAdd(&deg[dst[i]], 1.0f);
}

__global__ void rsqrt_k(float* __restrict__ p, int n) {
    int i = blockIdx.x * blockDim.x + threadIdx.x;
    if (i < n) p[i] = rsqrtf(p[i]);
}

__global__ void zero_k(float* __restrict__ p, int n) {
    int i = blockIdx.x * blockDim.x + threadIdx.x;
    if (i < n) p[i] = 0.0f;
}

__global__ void relu_k(float* __restrict__ p, int n) {
    int i = blockIdx.x * blockDim.x + threadIdx.x;
    if (i < n) p[i] = fmaxf(p[i], 0.0f);
}

// ---------------------------------------------------------------- WMMA GEMM
// C[M x 256] = A[M x K] * W[256 x K]^T   (W row-major [out_feat][k])
// Each wave computes a 16x64 strip: row tile (16 rows) x 4 column tiles.
// K is a compile-time constant -> guard-free main loop (only the K%4 tail,
// which exists solely for K=59, carries per-lane guards).
// MODE 0:  out2[m][n] = relu(acc + bias[n])                      (embed / FC)
// MODE 1:  out1[m][n] = acc;  out2[m][n] = dinv[m]^2*acc+bias[n] (conv pre-agg)
template <int K, int MODE>
__global__ void wmma_gemm_nt_k(const float* __restrict__ A,
                               const float* __restrict__ W,
                               const float* __restrict__ bias,
                               const float* __restrict__ dinv,
                               float* __restrict__ out1,
                               float* __restrict__ out2,
                               int M) {
    const int gwave = (blockIdx.x * blockDim.x + threadIdx.x) >> 5;
    const int lane  = threadIdx.x & 31;
    const int row_tile = gwave >> 2;        // which 16-row tile
    const int colg     = gwave & 3;         // which 64-col group
    if (row_tile * 16 >= M) return;
    const int row0 = row_tile * 16;
    const int n0   = colg * 64;

    const int lhalf = lane & 15;            // row (A) / col (B,D) within tile
    const int khi   = (lane >> 4) << 1;     // lanes 16-31 carry K+2,K+3

    const float* __restrict__ arow = A + (size_t)(row0 + lhalf) * K + khi;
    const float* __restrict__ wrow0 = W + (size_t)(n0 + lhalf) * K + khi;

    v8f acc0 = {}, acc1 = {}, acc2 = {}, acc3 = {};

    constexpr int K4 = K & ~3;              // guard-free portion

    #pragma unroll 4
    for (int k0 = 0; k0 < K4; k0 += 4) {
        // ---- A fragment: 16x4 f32, lane(lhalf) = row, (kk,kk+1) = K pair
        const v2f a = *(const v2f*)(arow + k0);
        // ---- B fragments: 4x16 f32 per column tile, B[k][n] = W[n][k]
        const v2f b0 = *(const v2f*)(wrow0 + k0);
        const v2f b1 = *(const v2f*)(wrow0 + (size_t)16 * K + k0);
        const v2f b2 = *(const v2f*)(wrow0 + (size_t)32 * K + k0);
        const v2f b3 = *(const v2f*)(wrow0 + (size_t)48 * K + k0);
        acc0 = __builtin_amdgcn_wmma_f32_16x16x4_f32(false, a, false, b0, (short)0, acc0, false, false);
        acc1 = __builtin_amdgcn_wmma_f32_16x16x4_f32(false, a, false, b1, (short)0, acc1, false, false);
        acc2 = __builtin_amdgcn_wmma_f32_16x16x4_f32(false, a, false, b2, (short)0, acc2, false, false);
        acc3 = __builtin_amdgcn_wmma_f32_16x16x4_f32(false, a, false, b3, (short)0, acc3, false, false);
    }

    if constexpr (K4 < K) {                 // tail (embed K=59: k=56..58)
        const int kk = K4 + khi;
        v2f a;
        a.x = (kk     < K) ? arow[K4]     : 0.0f;
        a.y = (kk + 1 < K) ? arow[K4 + 1] : 0.0f;
        #pragma unroll
        for (int t = 0; t < 4; ++t) {
            const float* wp = wrow0 + (size_t)(t * 16) * K + K4;
            v2f b;
            b.x = (kk     < K) ? wp[0] : 0.0f;
            b.y = (kk + 1 < K) ? wp[1] : 0.0f;
            v8f c = (t == 0) ? acc0 : (t == 1) ? acc1 : (t == 2) ? acc2 : acc3;
            c = __builtin_amdgcn_wmma_f32_16x16x4_f32(false, a, false, b, (short)0, c, false, false);
            if      (t == 0) acc0 = c;
            else if (t == 1) acc1 = c;
            else if (t == 2) acc2 = c;
            else             acc3 = c;
        }
    }

    // ---- epilogue: VGPR j, lanes 0-15 -> (M=j,N=lhalf); lanes 16-31 -> (M=j+8)
    const int mofs = (lane >> 4) * 8;
    #pragma unroll
    for (int t = 0; t < 4; ++t) {
        const v8f acc = (t == 0) ? acc0 : (t == 1) ? acc1 : (t == 2) ? acc2 : acc3;
        const int ncol = n0 + t * 16 + lhalf;
        const float bn = bias[ncol];
        #pragma unroll
        for (int j = 0; j < 8; ++j) {
            const int m = row0 + mofs + j;
            const float v = acc[j];
            const size_t o = (size_t)m * HDIM + ncol;
            if constexpr (MODE == 0) {
                out2[o] = fmaxf(v + bn, 0.0f);
            } else {
                out1[o] = v;
                const float di = dinv[m];
                out2[o] = di * di * v + bn;
            }
        }
    }
}

// ---------------------------------------------------------------- edge scatter
// one thread per (edge, 4-float chunk): agg[dst] += hw[src] * dinv[s]*dinv[d]
__global__ void scatter_edges_k(const float* __restrict__ hw,
                                const int* __restrict__ src,
                                const int* __restrict__ dst,
                                const float* __restrict__ dinv,
                                float* __restrict__ agg, int e) {
    const int idx = blockIdx.x * blockDim.x + threadIdx.x;  // < E*64 = 51.2M
    if (idx >= e * 64) return;
    const int ed = idx >> 6;
    const int c  = (idx & 63) << 2;
    const int s = src[ed], d = dst[ed];
    const float nrm = dinv[s] * dinv[d];
    const float4 v = *(const float4*)(hw + (size_t)s * HDIM + c);
    float* o = agg + (size_t)d * HDIM + c;
    atomicAdd(o + 0, v.x * nrm);
    atomicAdd(o + 1, v.y * nrm);
    atomicAdd(o + 2, v.z * nrm);
    atomicAdd(o + 3, v.w * nrm);
}

// ---------------------------------------------------------------- pooling
__global__ void pool_k(const float* __restrict__ h, const int* __restrict__ batch,
                       float* __restrict__ g, int n) {
    const int idx = blockIdx.x * blockDim.x + threadIdx.x;  // < N*64
    if (idx >= n * 64) return;
    const int node = idx >> 6;
    const int c    = (idx & 63) << 2;
    const int b = batch[node];
    const float4 v = *(const float4*)(h + (size_t)node * HDIM + c);
    float* o = g + (size_t)b * HDIM + c;
    atomicAdd(o + 0, v.x);
    atomicAdd(o + 1, v.y);
    atomicAdd(o + 2, v.z);
    atomicAdd(o + 3, v.w);
}

// ---------------------------------------------------------------- head
__global__ void head_k(const float* __restrict__ g, const float* __restrict__ ow,
                       const float* __restrict__ ob, float* __restrict__ out, int ng) {
    const int gi = blockIdx.x * blockDim.x + threadIdx.x;
    if (gi >= ng) return;
    float z0 = ob[0], z1 = ob[1];
    const float* row = g + (size_t)gi * HDIM;
    for (int k = 0; k < HDIM; k += 4) {
        const float4 v = *(const float4*)(row + k);
        z0 += v.x * ow[k] + v.y * ow[k + 1] + v.z * ow[k + 2] + v.w * ow[k + 3];
        z1 += v.x * ow[HDIM + k] + v.y * ow[HDIM + k + 1]
            + v.z * ow[HDIM + k + 2] + v.w * ow[HDIM + k + 3];
    }
    const float m = fmaxf(z0, z1);
    const float lse = m + logf(expf(z0 - m) + expf(z1 - m));
    out[gi * 2 + 0] = z0 - lse;
    out[gi * 2 + 1] = z1 - lse;
}

// ---------------------------------------------------------------- launch
static inline int cdiv(long long a, long long b) { return (int)((a + b - 1) / b); }

extern "C" void kernel_launch(void* const* d_in, const int* in_sizes, int n_in,
                              void* d_out, int out_size, void* d_ws, size_t ws_size,
                              hipStream_t stream) {
    (void)in_sizes; (void)n_in; (void)out_size; (void)ws_size;

    const float* x       = (const float*)d_in[0];
    const int*   eidx    = (const int*)d_in[1];     // [2,E] flat
    const int*   batch   = (const int*)d_in[2];
    const float* W_embed = (const float*)d_in[3];   // [256,59]
    const float* b_embed = (const float*)d_in[4];
    const float* conv_w  = (const float*)d_in[5];   // [3,256,256]
    const float* conv_b  = (const float*)d_in[6];   // [3,256]
    const float* fc_w    = (const float*)d_in[7];   // [3,256,256]
    const float* fc_b    = (const float*)d_in[8];   // [3,256]
    const float* out_w   = (const float*)d_in[9];   // [2,256]
    const float* out_b   = (const float*)d_in[10];  // [2]
    float* out = (float*)d_out;

    const int* src = eidx;
    const int* dst = eidx + N_EDGES;

    // ---- workspace layout (256B aligned slices)
    char* ws = (char*)d_ws;
    size_t off = 0;
    float* dinv = (float*)(ws + off); off += ((size_t)N_NODES * 4 + 255) & ~(size_t)255;
    const size_t nodeBuf = (size_t)N_NODES * HDIM * 4;   // 51.2 MB, 256B multiple
    float* buf[3];
    buf[0] = (float*)(ws + off); off += nodeBuf;
    buf[1] = (float*)(ws + off); off += nodeBuf;
    buf[2] = (float*)(ws + off); off += nodeBuf;
    const size_t gBuf = (size_t)N_GRAPHS * HDIM * 4;     // 2 MB each
    float* g0 = (float*)(ws + off); off += gBuf;
    float* g1 = (float*)(ws + off); off += gBuf;

    const int TB = 256;

    // ---- 1. degree / dinv
    deg_init_k<<<cdiv(N_NODES, TB), TB, 0, stream>>>(dinv, N_NODES);
    deg_count_k<<<cdiv(N_EDGES, TB), TB, 0, stream>>>(dst, dinv, N_EDGES);
    rsqrt_k<<<cdiv(N_NODES, TB), TB, 0, stream>>>(dinv, N_NODES);

    // GEMM launch geometry: (M/16 tiles)*4 col-groups waves, 32 threads each
    const int wavesN = (N_NODES / 16) * 4;               // 12500
    const int gemmNBlocks = cdiv((long long)wavesN * 32, TB);
    const int wavesG = (N_GRAPHS / 16) * 4;              // 512
    const int gemmGBlocks = cdiv((long long)wavesG * 32, TB);

    // ---- 2. embed: h0 = relu(x @ W_embed^T + b_embed)   (K=59, guarded tail)
    wmma_gemm_nt_k<IN_FEATS, 0><<<gemmNBlocks, TB, 0, stream>>>(
        x, W_embed, b_embed, nullptr, nullptr, buf[0], N_NODES);

    // ---- 3. GCN convs
    int hi = 0;
    for (int l = 0; l < L_CONV; ++l) {
        float* h   = buf[hi];
        float* hw  = buf[(hi + 1) % 3];
        float* agg = buf[(hi + 2) % 3];
        // hw = h @ w^T ; agg = dinv^2*hw + bias
        wmma_gemm_nt_k<HDIM, 1><<<gemmNBlocks, TB, 0, stream>>>(
            h, conv_w + (size_t)l * HDIM * HDIM, conv_b + (size_t)l * HDIM,
            dinv, hw, agg, N_NODES);
        // agg[dst] += hw[src]*dinv[src]*dinv[dst]
        scatter_edges_k<<<cdiv((long long)N_EDGES * 64, TB), TB, 0, stream>>>(
            hw, src, dst, dinv, agg, N_EDGES);
        // relu
        relu_k<<<cdiv((long long)N_NODES * HDIM, TB), TB, 0, stream>>>(
            agg, N_NODES * HDIM);
        hi = (hi + 2) % 3;
    }
    float* hFinal = buf[hi];

    // ---- 4. global add pool
    zero_k<<<cdiv((long long)N_GRAPHS * HDIM, TB), TB, 0, stream>>>(g0, N_GRAPHS * HDIM);
    pool_k<<<cdiv((long long)N_NODES * 64, TB), TB, 0, stream>>>(hFinal, batch, g0, N_NODES);

    // ---- 5. FC layers: g = relu(g @ fc_w^T + fc_b)
    float* gin = g0;
    float* gout = g1;
    for (int l = 0; l < L_FC; ++l) {
        wmma_gemm_nt_k<HDIM, 0><<<gemmGBlocks, TB, 0, stream>>>(
            gin, fc_w + (size_t)l * HDIM * HDIM, fc_b + (size_t)l * HDIM,
            nullptr, nullptr, gout, N_GRAPHS);
        float* t = gin; gin = gout; gout = t;
    }

    // ---- 6. logits + log_softmax
    head_k<<<cdiv(N_GRAPHS, TB), TB, 0, stream>>>(gin, out_w, out_b, out, N_GRAPHS);
}